// RTDetrHungarianMatcher_11467562680412
// MI455X (gfx1250) — compile-verified
//
#include <hip/hip_runtime.h>
#include <cstdint>
#include <cstddef>

// ---------------- problem constants (from reference) ----------------
#define ALPHA_     0.25f
#define BBOX_COST  5.0f
#define CLASS_COST 2.0f
#define GIOU_COST  2.0f
#define EPS_       1e-6f
#define LOG_EPS_   1e-8f

#define B_  32
#define Q_  300
#define C_  80
#define M_  3200
#define N_  (B_ * Q_)      // 9600

#define TN  16             // rows (n) per block
#define TM  256            // cols (m) per block == blockDim.x

// ---------------- CDNA5 async global->LDS helpers ----------------
typedef int v4i_t __attribute__((vector_size(4 * sizeof(int))));
typedef __attribute__((address_space(1))) v4i_t glob_v4i_t;
typedef __attribute__((address_space(3))) v4i_t lds_v4i_t;
typedef __attribute__((address_space(3))) void  lds_void_t;

#if __has_builtin(__builtin_amdgcn_global_load_async_to_lds_b128)
#define HAVE_ASYNC_BUILTIN 1
#endif

__device__ __forceinline__ void async_copy_b128(const void* g, void* l) {
#ifdef HAVE_ASYNC_BUILTIN
  __builtin_amdgcn_global_load_async_to_lds_b128(
      (glob_v4i_t*)g, (lds_v4i_t*)l, 0, 0);
#else
  // VDST = VGPR holding LDS byte offset, VADDR = 64-bit global address
  // (CDNA5 ISA: GLOBAL_LOAD_ASYNC_TO_LDS_B128, GV mode)
  uint32_t lds_off = (uint32_t)(uintptr_t)(lds_void_t*)l;
  asm volatile("global_load_async_to_lds_b128 %0, %1, off"
               :: "v"(lds_off), "v"(g) : "memory");
#endif
}

__device__ __forceinline__ void wait_async0() {
#if __has_builtin(__builtin_amdgcn_s_wait_asynccnt)
  __builtin_amdgcn_s_wait_asynccnt(0);
#else
  asm volatile("s_wait_asynccnt 0" ::: "memory");
#endif
}

// ---------------- kernel 1: per-(n,class) focal class-cost table ----------------
// table[n][c] = ALPHA*(1-p)^2*(-log(p+eps)) - (1-ALPHA)*p^2*(-log(1-p+eps))
__global__ __launch_bounds__(256)
void class_cost_kernel(const float* __restrict__ logits, float* __restrict__ table) {
  int i = blockIdx.x * 256 + threadIdx.x;
  if (i >= N_ * C_) return;
  float x  = logits[i];
  // fast sigmoid: v_exp_f32 + v_rcp_f32 (avoid the IEEE div_scale/div_fixup chain)
  float p  = __builtin_amdgcn_rcpf(1.0f + __expf(-x));
  float om = 1.0f - p;
  float pos = ALPHA_ * om * om * (-__logf(p + LOG_EPS_));
  float neg = (1.0f - ALPHA_) * p * p * (-__logf(om + LOG_EPS_));
  table[i] = pos - neg;
}

// ---------------- kernel 2: per-n pred box aux {cx,cy,w,h, x0,y0,x1,y1} ----------------
__global__ __launch_bounds__(256)
void pred_aux_kernel(const float* __restrict__ pred, float* __restrict__ aux) {
  int n = blockIdx.x * 256 + threadIdx.x;
  if (n >= N_) return;
  float4 b = ((const float4*)pred)[n];
  float hw = 0.5f * b.z, hh = 0.5f * b.w;
  ((float4*)aux)[2 * n]     = b;                                            // cx,cy,w,h
  ((float4*)aux)[2 * n + 1] = make_float4(b.x - hw, b.y - hh, b.x + hw, b.y + hh); // x0,y0,x1,y1
}

// ---------------- kernel 3: main N x M cost matrix ----------------
__global__ __launch_bounds__(TM)
void cost_kernel(const float* __restrict__ classTable,  // [N_][C_]
                 const float* __restrict__ predAux,     // [N_][8]
                 const float* __restrict__ tboxes,      // [M_][4] cxcywh
                 const int*   __restrict__ tids,        // [M_]
                 float*       __restrict__ out) {       // [N_][M_]
  __shared__ __align__(16) float sClass[TN * C_];       // 5 KB

  const int tid = threadIdx.x;
  const int n0  = blockIdx.y * TN;
  const int m   = blockIdx.x * TM + tid;

  // Stage 16 contiguous class-table rows (5120 B) into LDS with async B128 copies.
  {
    const char* src = (const char*)(classTable + (size_t)n0 * C_);
    char*       dst = (char*)sClass;
    for (int off = tid * 16; off < TN * C_ * 4; off += TM * 16)
      async_copy_b128(src + off, dst + off);
  }
  wait_async0();
  __syncthreads();

  if (m < M_) {
    float4 tb  = ((const float4*)tboxes)[m];
    int    cid = tids[m];
    float thw = 0.5f * tb.z, thh = 0.5f * tb.w;
    float tx0 = tb.x - thw, ty0 = tb.y - thh;
    float tx1 = tb.x + thw, ty1 = tb.y + thh;
    float tarea = tb.z * tb.w;
    size_t obase = (size_t)n0 * M_ + (size_t)m;

#pragma unroll 4
    for (int nl = 0; nl < TN; ++nl) {
      // Uniform address across the block -> scalar (SMEM) loads.
      const float4* pa = (const float4*)(predAux + (size_t)(n0 + nl) * 8);
      float4 p0 = pa[0];   // cx,cy,w,h
      float4 p1 = pa[1];   // x0,y0,x1,y1

      // L1 in cxcywh space
      float l1 = fabsf(p0.x - tb.x) + fabsf(p0.y - tb.y) +
                 fabsf(p0.z - tb.z) + fabsf(p0.w - tb.w);

      // GIoU in xyxy space
      float parea = p0.z * p0.w;
      float ix0 = fmaxf(p1.x, tx0), iy0 = fmaxf(p1.y, ty0);
      float ix1 = fminf(p1.z, tx1), iy1 = fminf(p1.w, ty1);
      float iw = fmaxf(ix1 - ix0, 0.0f), ih = fmaxf(iy1 - iy0, 0.0f);
      float inter = iw * ih;
      float uni   = parea + tarea - inter;
      float ex0 = fminf(p1.x, tx0), ey0 = fminf(p1.y, ty0);
      float ex1 = fmaxf(p1.z, tx1), ey1 = fmaxf(p1.w, ty1);
      float ew = fmaxf(ex1 - ex0, 0.0f), eh = fmaxf(ey1 - ey0, 0.0f);
      float enc = ew * eh;
      float giou = inter * __builtin_amdgcn_rcpf(uni + EPS_)
                 - (enc - uni) * __builtin_amdgcn_rcpf(enc + EPS_);

      float cls = sClass[nl * C_ + cid];   // LDS gather by class id

      out[obase + (size_t)nl * M_] =
          BBOX_COST * l1 + CLASS_COST * cls - GIOU_COST * giou;
    }
  }
}

// ---------------- host entry ----------------
extern "C" void kernel_launch(void* const* d_in, const int* in_sizes, int n_in,
                              void* d_out, int out_size, void* d_ws, size_t ws_size,
                              hipStream_t stream) {
  (void)in_sizes; (void)n_in; (void)out_size; (void)ws_size;

  const float* logits = (const float*)d_in[0];   // (B,Q,C) f32
  const float* pred   = (const float*)d_in[1];   // (B,Q,4) f32
  const int*   tids   = (const int*)d_in[2];     // (M,)    i32
  const float* tbox   = (const float*)d_in[3];   // (M,4)   f32
  float* out = (float*)d_out;                    // (B,Q,M) f32

  // workspace: class table (N*C floats = 3.0 MB) then pred aux (N*8 floats)
  float* classTable = (float*)d_ws;
  float* predAux    = classTable + (size_t)N_ * C_;

  class_cost_kernel<<<(N_ * C_ + 255) / 256, 256, 0, stream>>>(logits, classTable);
  pred_aux_kernel  <<<(N_ + 255) / 256,      256, 0, stream>>>(pred, predAux);

  dim3 grid((M_ + TM - 1) / TM, N_ / TN);
  cost_kernel<<<grid, TM, 0, stream>>>(classTable, predAux, tbox, tids, out);
}